// DynamicDownsampling_44392781971546
// MI455X (gfx1250) — compile-verified
//
#include <hip/hip_runtime.h>
#include <hip/hip_bf16.h>
#include <stdint.h>

// Problem constants (from reference)
#define BATCH 4
#define CH    32
#define HH    256
#define WW    256
#define KS    5
#define KK    25          // KS*KS
#define PADR  2           // (KS-1)/2

// Tiling: 32x16 pixel tile, 256 threads, each thread owns a 1x2 pixel column
// (vertically adjacent outputs share 4 of 5 patch rows -> 15 LDS reads/output)
#define TILE_W 32
#define TILE_H 16
#define ROWS_PT 2                  // rows per thread
#define TPB    256                 // 8 waves (wave32)
#define LW (TILE_W + 2*PADR)       // 36
#define LH (TILE_H + 2*PADR)       // 20
#define LSZ (LH * LW)              // 720 floats per tile buffer

typedef __attribute__((address_space(1))) int gbl_int_t;   // "__device__ int*" in HIP diagnostics
typedef __attribute__((address_space(3))) int lds_int_t;   // "__shared__ int*"

// ---- CDNA5 async global->LDS copy (ASYNCcnt-tracked) -----------------------
__device__ __forceinline__ void async_copy_f32(const float* gptr, float* lptr) {
#if __has_builtin(__builtin_amdgcn_global_load_async_to_lds_b32)
    __builtin_amdgcn_global_load_async_to_lds_b32(
        (gbl_int_t*)(size_t)(const void*)gptr,
        (lds_int_t*)(unsigned)(size_t)(void*)lptr,  // low 32b of generic LDS addr = LDS byte offset
        0, 0);
#else
    unsigned lds_b = (unsigned)(size_t)(void*)lptr;
    unsigned long long ga = (unsigned long long)(size_t)(const void*)gptr;
    asm volatile("global_load_async_to_lds_b32 %0, %1, off"
                 :: "v"(lds_b), "v"(ga) : "memory");
#endif
}

__device__ __forceinline__ void wait_async0() {
#if __has_builtin(__builtin_amdgcn_s_wait_asynccnt)
    __builtin_amdgcn_s_wait_asynccnt(0);
#else
    asm volatile("s_wait_asynccnt 0x0" ::: "memory");
#endif
}

// Issue the (TILE+halo) tile of one channel plane as async loads into LDS.
// Replicate ("edge") padding handled by clamping source coordinates.
__device__ __forceinline__ void issue_tile_load(const float* plane, float* dst,
                                                int tid, int y0, int x0) {
#pragma unroll
    for (int base = 0; base < LSZ; base += TPB) {     // 0, 256, 512
        int i = base + tid;
        if (base + TPB <= LSZ || i < LSZ) {           // last chunk is partial (720-512=208)
            int r = i / LW, cc = i - r * LW;
            int gh = min(max(y0 + r - PADR, 0), HH - 1);
            int gw = min(max(x0 + cc - PADR, 0), WW - 1);
            async_copy_f32(plane + (size_t)gh * WW + gw, dst + i);
        }
    }
}

// Per-pixel softmax over 25 logits, kept in registers.
__device__ __forceinline__ void softmax25(const float* kb, float* w) {
    float m = -3.402823466e38f;
#pragma unroll
    for (int p = 0; p < KK; ++p) {
        w[p] = kb[(size_t)p * (HH * WW)];   // lanes contiguous in w -> coalesced
        m = fmaxf(m, w[p]);
    }
    float s = 0.f;
#pragma unroll
    for (int p = 0; p < KK; ++p) { w[p] = __expf(w[p] - m); s += w[p]; }
    const float inv = 1.0f / s;
#pragma unroll
    for (int p = 0; p < KK; ++p) w[p] *= inv;
}

__global__ __launch_bounds__(TPB)
void DynamicDownsampling_kernel(const float* __restrict__ x,
                                const float* __restrict__ kern,
                                float* __restrict__ out) {
    __shared__ float tile[2][LSZ];

    const int tid = threadIdx.x;
    const int tx  = tid & (TILE_W - 1);
    const int tyq = tid >> 5;                 // 0..7 (wave = one output-row pair)
    const int ly0 = tyq * ROWS_PT;            // first tile-local output row
    const int x0  = blockIdx.x * TILE_W;
    const int y0  = blockIdx.y * TILE_H;
    const int b   = blockIdx.z;
    const int px  = x0 + tx;
    const int py0 = y0 + ly0;                 // second row is py0+1

    // ---- softmax weights for both owned pixels (registers) -----------------
    float w0[KK], w1[KK];
    const float* kb = kern + (size_t)b * KK * HH * WW + (size_t)py0 * WW + px;
    softmax25(kb,      w0);
    softmax25(kb + WW, w1);

    const float* xb = x   + (size_t)b * CH * HH * WW;
    float*       ob = out + (size_t)b * CH * HH * WW
                          + (size_t)py0 * WW + px;

    // ---- channel loop: async double-buffered x tiles through LDS -----------
    issue_tile_load(xb, tile[0], tid, y0, x0);          // prefetch c = 0

    for (int c = 0; c < CH; ++c) {
        wait_async0();          // my wave's outstanding tile loads complete
        __syncthreads();        // all waves' portions of tile[c&1] visible;
                                // also: all reads of tile[(c+1)&1] (iter c-1) done

        if (c + 1 < CH)         // overlap next channel's load with compute
            issue_tile_load(xb + (size_t)(c + 1) * (HH * WW),
                            tile[(c + 1) & 1], tid, y0, x0);

        const float* tb = tile[c & 1];
        float acc0 = 0.f, acc1 = 0.f;
        // 6 shared rows cover both 5x5 patches: row r feeds acc0 (r<5, w0[r])
        // and acc1 (r>=1, w1[r-1]) -> 30 DS loads for 2 outputs.
#pragma unroll
        for (int r = 0; r < KS + 1; ++r) {
            float v[KS];
            const int rowbase = (ly0 + r) * LW + tx;
#pragma unroll
            for (int dj = 0; dj < KS; ++dj) v[dj] = tb[rowbase + dj];
            if (r < KS) {
#pragma unroll
                for (int dj = 0; dj < KS; ++dj)
                    acc0 = fmaf(v[dj], w0[r * KS + dj], acc0);
            }
            if (r >= 1) {
#pragma unroll
                for (int dj = 0; dj < KS; ++dj)
                    acc1 = fmaf(v[dj], w1[(r - 1) * KS + dj], acc1);
            }
        }
        ob[(size_t)c * (HH * WW)]      = acc0;   // coalesced (lanes contiguous in x)
        ob[(size_t)c * (HH * WW) + WW] = acc1;
    }
}

extern "C" void kernel_launch(void* const* d_in, const int* in_sizes, int n_in,
                              void* d_out, int out_size, void* d_ws, size_t ws_size,
                              hipStream_t stream) {
    const float* x    = (const float*)d_in[0];  // [4,32,256,256]
    const float* kern = (const float*)d_in[1];  // [4,25,256,256]
    float* out        = (float*)d_out;          // [4,32,256,256]

    dim3 grid(WW / TILE_W, HH / TILE_H, BATCH); // (8, 16, 4)
    DynamicDownsampling_kernel<<<grid, TPB, 0, stream>>>(x, kern, out);
}